// RT60Estimator_54056458387805
// MI455X (gfx1250) — compile-verified
//
#include <hip/hip_runtime.h>
#include <hip/hip_bf16.h>
#include <math.h>

// ---------------- problem constants (resolved from the reference) ----------
#define NS        960000      // input samples
#define SRATE     8000
#define WINL      400
#define HOPL      300
#define NFFTC     512
#define KB        257         // K_BINS = NFFT/2+1
#define TF        3201        // frames
#define SF        3200        // S = T-1
#define LMAXC     26          // LMAX
#define NFREQS    4000        // output length
#define FT        0.0375f     // HOP/SR
#define BIGF      1.0e9f

// GEMM padding
#define MP        3216        // 201*16  (>= TF)
#define KA        400         // reduction dim (window length)
#define NP        272         // 17*16   (>= KB)

typedef __attribute__((ext_vector_type(2))) float v2f;
typedef __attribute__((ext_vector_type(8))) float v8f;

#if defined(__has_builtin)
#if __has_builtin(__builtin_amdgcn_wmma_f32_16x16x4_f32)
#define USE_WMMA_F32 1
#endif
#endif
#ifndef USE_WMMA_F32
#define USE_WMMA_F32 0
#endif

// ---------------- stage 0a: build windowed-frame matrix A (MP x KA) --------
__global__ __launch_bounds__(256) void rt60_build_A(const float* __restrict__ x,
                                                    float* __restrict__ A) {
  int idx = blockIdx.x * 256 + threadIdx.x;
  if (idx >= MP * KA) return;
  int m = idx / KA;
  int n = idx - m * KA;
  float v = 0.0f;
  if (m < TF) {
    // frame sample s = n+56 (Hann window support inside NFFT); reflect pad 256
    int j = m * HOPL + n - 200;
    if (j < 0) j = -j;
    if (j >= NS) j = 2 * NS - 2 - j;
    float w = 0.5f - 0.5f * cosf(6.28318530717958647692f * (float)n / 400.0f);
    v = x[j] * w;
  }
  A[idx] = v;
}

// ---------------- stage 0b: DFT basis matrices Bc/Bs (KA x NP) -------------
__global__ __launch_bounds__(256) void rt60_build_B(float* __restrict__ Bc,
                                                    float* __restrict__ Bs) {
  int idx = blockIdx.x * 256 + threadIdx.x;
  if (idx >= KA * NP) return;
  int ka = idx / NP;     // sample index within window (s = ka+56)
  int nb = idx - ka * NP; // frequency bin
  float c = 0.0f, s = 0.0f;
  if (nb < KB) {
    // exact periodic reduction: cos(2*pi*k*s/512) depends on (k*s) mod 512
    int q = (nb * (ka + 56)) & 511;
    double ang = 6.283185307179586476925286766559 * (double)q / 512.0;
    c = (float)cos(ang);
    s = (float)sin(ang);
  }
  Bc[idx] = c;
  Bs[idx] = s;
}

// ---------------- stage 1: fp32 WMMA GEMM -> power spectrogram -------------
// One wave per 16x16 output tile. D = A(16xK) * B(Kx16), K-step = 4.
// Epilogue fuses power = Re^2 + Im^2 and writes P transposed: P[k][t].
__global__ __launch_bounds__(32) void rt60_gemm_power(const float* __restrict__ A,
                                                      const float* __restrict__ Bc,
                                                      const float* __restrict__ Bs,
                                                      float* __restrict__ P) {
  const int lane = threadIdx.x;
  const int lm = lane & 15;   // A row / B-D column within tile
  const int lh = lane >> 4;   // lane-half selects K pair / M half
  const int tm = blockIdx.x;  // M tile (frames)
  const int tn = blockIdx.y;  // N tile (bins)

  const float* __restrict__ arow = A + (size_t)(tm * 16 + lm) * KA;
  const float* __restrict__ bcp  = Bc + tn * 16 + lm;
  const float* __restrict__ bsp  = Bs + tn * 16 + lm;

  v8f accR = {};
  v8f accI = {};

  for (int k0 = 0; k0 < KA; k0 += 4) {
    const int kr = k0 + 2 * lh;
    __builtin_prefetch(arow + k0 + 32, 0, 0);  // -> global_prefetch_b8
    v2f a, bc, bs;
    a.x  = arow[kr];
    a.y  = arow[kr + 1];
    bc.x = bcp[(size_t)kr * NP];
    bc.y = bcp[(size_t)(kr + 1) * NP];
    bs.x = bsp[(size_t)kr * NP];
    bs.y = bsp[(size_t)(kr + 1) * NP];
#if USE_WMMA_F32
    accR = __builtin_amdgcn_wmma_f32_16x16x4_f32(false, a, false, bc,
                                                 (short)0, accR, false, false);
    accI = __builtin_amdgcn_wmma_f32_16x16x4_f32(false, a, false, bs,
                                                 (short)0, accI, false, false);
#else
    // shuffle-based fallback (compile-safety only)
    #pragma unroll
    for (int v = 0; v < 8; ++v) {
      int m = 8 * lh + v;
      #pragma unroll
      for (int kq = 0; kq < 4; ++kq) {
        int srcA = ((kq >> 1) << 4) + m;
        int srcB = ((kq >> 1) << 4) + lm;
        float av = __shfl((kq & 1) ? a.y : a.x, srcA, 32);
        accR[v] += av * __shfl((kq & 1) ? bc.y : bc.x, srcB, 32);
        accI[v] += av * __shfl((kq & 1) ? bs.y : bs.x, srcB, 32);
      }
    }
#endif
  }

  #pragma unroll
  for (int v = 0; v < 8; ++v) {
    int t = tm * 16 + 8 * lh + v;   // frame index (D row)
    int k = tn * 16 + lm;           // bin index   (D col)
    if (t < TF && k < KB) {
      float re = accR[v], im = accI[v];
      P[(size_t)k * TF + t] = re * re + im * im;
    }
  }
}

// ---------------- stage 2: per-bin decay analysis + median -----------------
__global__ __launch_bounds__(256) void rt60_perbin(const float* __restrict__ P,
                                                   float* __restrict__ med0,
                                                   int* __restrict__ proc) {
  __shared__ float pr[TF + 3];
  __shared__ int   rfv[4096];
  __shared__ float sv[4096];
  __shared__ int   red[256];
  __shared__ int   sMaxRun, sMcount;

  const int tid = threadIdx.x;
  const int k   = blockIdx.x;
  const float* __restrict__ prow = P + (size_t)k * TF;

  for (int t = tid; t < TF; t += 256) pr[t] = prow[t];
  __syncthreads();

  // init suffix-min scan values: d[t] = (p[t+1] < p[t]); value = t if !d else INF
  for (int c = 0; c < 16; ++c) {
    int t = tid + 256 * c;
    if (t < SF) rfv[t] = (pr[t + 1] < pr[t]) ? (1 << 30) : t;
    else        rfv[t] = SF;
  }
  __syncthreads();

  // Hillis-Steele suffix-min over 4096 entries
  for (int off = 1; off < 4096; off <<= 1) {
    int tmp[16];
    #pragma unroll
    for (int c = 0; c < 16; ++c) {
      int t = tid + 256 * c;
      int u = t + off;
      tmp[c] = (u < 4096) ? rfv[u] : (1 << 30);
    }
    __syncthreads();
    #pragma unroll
    for (int c = 0; c < 16; ++c) {
      int t = tid + 256 * c;
      if (tmp[c] < rfv[t]) rfv[t] = tmp[c];
    }
    __syncthreads();
  }

  // rf[t] = (next non-decreasing index) - t ; track max run
  int lmax = 0;
  for (int c = 0; c < 13; ++c) {
    int t = tid + 256 * c;
    if (t < SF) {
      int rf = rfv[t] - t;
      rfv[t] = rf;
      if (rf > lmax) lmax = rf;
    }
  }
  red[tid] = lmax;
  __syncthreads();
  for (int s = 128; s > 0; s >>= 1) {
    if (tid < s) red[tid] = max(red[tid], red[tid + s]);
    __syncthreads();
  }
  if (tid == 0) sMaxRun = red[0];
  __syncthreads();

  const int  maxrun    = sMaxRun;
  const int  Lf        = min(maxrun + 1, LMAXC);
  const bool processed = (maxrun >= 2);   // MIN_W - 1

  // per-frame EDC / regression / RT
  int zcount = 0;
  #pragma unroll 1
  for (int c = 0; c < 16; ++c) {
    int t = tid + 256 * c;
    if (t < SF) {
      float edc[LMAXC];
      float ylog[LMAXC];
      float acc = 0.0f;
      #pragma unroll
      for (int l = LMAXC - 1; l >= 0; --l) {
        if (l < Lf) {
          int g = t + l; if (g > SF) g = SF;
          acc += pr[g];
          edc[l] = acc;
        }
      }
      float edc0 = fmaxf(edc[0], 1e-30f);
      int idx10 = 0;
      #pragma unroll
      for (int l = 0; l < LMAXC; ++l) {
        if (l < Lf) {
          float y = 10.0f * log10f(edc[l] / edc0);
          ylog[l] = y;
          if (-y < 10.0f) idx10++;
        }
      }
      float Sw = 0, Sx = 0, Sy = 0, Sxy = 0, Sxx = 0;
      #pragma unroll
      for (int l = 0; l < LMAXC; ++l) {
        if (l < Lf && l >= idx10) {
          float xf = (float)l, y = ylog[l];
          Sw += 1.0f; Sx += xf; Sy += y; Sxy += xf * y; Sxx += xf * xf;
        }
      }
      float mm    = fmaxf(Sw, 1.0f);
      float num   = Sxy - Sx * Sy / mm;
      float den   = fmaxf(Sxx - Sx * Sx / mm, 1e-12f);
      float slope = num / den;
      int n_reg   = Lf - idx10;
      float rt;
      if (n_reg < 2)            rt = (float)idx10 * FT;
      else if (slope < 1e-8f)   rt = (float)Lf * FT;
      else                      rt = (-60.0f / slope) * FT;
      bool zone = processed && (rfv[t] >= Lf - 1);
      sv[t] = zone ? rt : BIGF;
      if (zone) zcount++;
    } else {
      sv[t] = BIGF;
    }
  }
  __syncthreads();

  red[tid] = zcount;
  __syncthreads();
  for (int s = 128; s > 0; s >>= 1) {
    if (tid < s) red[tid] += red[tid + s];
    __syncthreads();
  }
  if (tid == 0) sMcount = red[0];
  __syncthreads();

  // bitonic sort of sv[0..4095] ascending
  for (int kk = 2; kk <= 4096; kk <<= 1) {
    for (int j = kk >> 1; j > 0; j >>= 1) {
      #pragma unroll 1
      for (int c = 0; c < 16; ++c) {
        int t = tid + 256 * c;
        int partner = t ^ j;
        if (partner > t) {
          bool up = ((t & kk) == 0);
          float x0 = sv[t], x1 = sv[partner];
          if ((x0 > x1) == up) { sv[t] = x1; sv[partner] = x0; }
        }
      }
      __syncthreads();
    }
  }

  if (tid == 0) {
    int mcount = sMcount;
    int mi = (mcount - 1) / 2;
    if (mi < 0) mi = 0;
    if (mi > SF - 1) mi = SF - 1;
    med0[k] = processed ? sv[mi] : 0.0f;
    proc[k] = processed ? 1 : 0;
  }
}

// ---------------- stage 3: sequential neighbor fill (single thread) --------
// r[f] recurrence only ever reads: prev (already computed, kept in register),
// and ORIGINAL r0 values at f and f+1 (0 beyond KB).
__global__ void rt60_fill(const float* __restrict__ med0,
                          const int* __restrict__ proc,
                          float* __restrict__ r) {
  if (threadIdx.x != 0 || blockIdx.x != 0) return;
  float prev = 0.0f;
  for (int f = 0; f < NFREQS; ++f) {
    float cur0  = 0.0f, next0 = 0.0f;
    bool  unpr  = true;
    if (f < KB) {
      int p = proc[f];
      cur0 = p ? med0[f] : 0.0f;
      unpr = (p == 0);
    }
    if (f + 1 < KB) next0 = proc[f + 1] ? med0[f + 1] : 0.0f;
    float v;
    if (!unpr)                 v = cur0;
    else if (f == 0)           v = (cur0 + next0) * 0.5f;
    else if (f == NFREQS - 1)  v = (prev + cur0) * 0.5f;
    else                       v = (prev + cur0 + next0) * (1.0f / 3.0f);
    r[f] = v;
    prev = v;
  }
}

// ---------------- launch ---------------------------------------------------
extern "C" void kernel_launch(void* const* d_in, const int* in_sizes, int n_in,
                              void* d_out, int out_size, void* d_ws, size_t ws_size,
                              hipStream_t stream) {
  (void)in_sizes; (void)n_in; (void)out_size; (void)ws_size;
  const float* x = (const float*)d_in[0];
  float* r = (float*)d_out;

  float* A    = (float*)d_ws;                 // MP*KA        = 1,286,400 f
  float* Bc   = A  + (size_t)MP * KA;         // KA*NP        =   108,800 f
  float* Bs   = Bc + (size_t)KA * NP;         // KA*NP        =   108,800 f
  float* P    = Bs + (size_t)KA * NP;         // KB*TF        =   822,657 f
  float* med0 = P  + (size_t)KB * TF;         // KB
  int*   proc = (int*)(med0 + KB);            // KB           (~9.3 MB total)

  rt60_build_A<<<(MP * KA + 255) / 256, 256, 0, stream>>>(x, A);
  rt60_build_B<<<(KA * NP + 255) / 256, 256, 0, stream>>>(Bc, Bs);
  rt60_gemm_power<<<dim3(MP / 16, NP / 16), 32, 0, stream>>>(A, Bc, Bs, P);
  rt60_perbin<<<KB, 256, 0, stream>>>(P, med0, proc);
  rt60_fill<<<1, 1, 0, stream>>>(med0, proc, r);
}